// SHT_60498909331754
// MI455X (gfx1250) — compile-verified
//
#include <hip/hip_runtime.h>

// ---------------- static HEALPix geometry (nside = 64) ----------------
#define NSIDE    64
#define NPIX     49152
#define NLAT     255         // rings
#define NLON     256         // equatorial ring length
#define LMAX     255
#define MMAX     129
#define NEQ      129         // equatorial rings
#define EQ_START 8064        // first equatorial pixel
#define NCAP_R   63
#define LCAP     252         // padded cap ring length
#define BT       512         // batch * channels = 8*64
#define MPAD     132         // padded m for the c[] intermediate
#define FM       144         // padded m for the DFT table (9 m-tiles of 16)
#define RING_EQ0 63          // first equatorial ring index
#define RING_EQ1 192         // first south-cap ring index

#define PITCH64  73          // LDS pitch for 64-wide tiles (gcd(73,64)=1)
#define PITCH16  17          // LDS pitch for 16-wide tiles

#define TWO_PI_OVER_256 0.024543692606170260f

typedef __attribute__((ext_vector_type(2))) float v2f;
typedef __attribute__((ext_vector_type(8))) float v8f;

__device__ __forceinline__ v8f wmma_f32(v2f a, v2f b, v8f c) {
    // V_WMMA_F32_16X16X4_F32 : D = A(16x4) * B(4x16) + C(16x16), fp32
    return __builtin_amdgcn_wmma_f32_16x16x4_f32(
        /*neg_a=*/false, a, /*neg_b=*/false, b,
        /*c_mod=*/(short)0, c, /*reuse_a=*/false, /*reuse_b=*/false);
}

// ---------------------------------------------------------------------
// Kernel 0: DFT table F[m][j] = exp(-i 2pi m j / 256), layout [FM][NLON].
// ---------------------------------------------------------------------
__global__ void build_dft_kernel(float* __restrict__ Fre, float* __restrict__ Fim) {
    int idx = blockIdx.x * blockDim.x + threadIdx.x;   // m = idx>>8, j = idx&255
    int m = idx >> 8;
    int j = idx & 255;
    float ang = -TWO_PI_OVER_256 * (float)((m * j) & 255);
    float s, c;
    __sincosf(ang, &s, &c);
    Fre[idx] = c;
    Fim[idx] = s;
}

// ---------------------------------------------------------------------
// Phase 1: per-ring DFT -> c[k][m][b] (re/im planes).
// Block = 128 threads (4 waves) covering 64 b x 16 m for one ring:
// the gathered x tile is staged once, each wave consumes a 16-b slice.
// grid = (NLAT, BT/64, ceil(MMAX/16))
// ---------------------------------------------------------------------
__global__ void __launch_bounds__(128) sht_phase1(
    const float* __restrict__ x, const float* __restrict__ E_cap,
    const float* __restrict__ eq_phase, const int* __restrict__ cap_idx,
    const float* __restrict__ Fre, const float* __restrict__ Fim,
    float* __restrict__ c_re, float* __restrict__ c_im)
{
    const int k    = blockIdx.x;          // ring 0..254
    const int b0   = blockIdx.y * 64;
    const int m0   = blockIdx.z * 16;
    const int tid  = threadIdx.x;
    const int w    = tid >> 5;            // wave id 0..3
    const int lane = tid & 31;

    __shared__ float As [32][PITCH64];    // [j][b(64)]
    __shared__ float Bre[32][PITCH16];    // [j][m(16)]
    __shared__ float Bim[32][PITCH16];

    const bool is_eq = (k >= RING_EQ0) && (k < RING_EQ1);
    const int  cr    = (k < RING_EQ0) ? k : (k - MMAX);   // cap row (south: k-129)
    const int  jlen  = is_eq ? NLON : LCAP;

    v8f acc_re = {}; v8f acc_im = {};
    const int row   = lane & 15;
    const int koff0 = (lane < 16) ? 0 : 2;
    const int wb    = w * 16;             // wave's b offset inside the tile

    for (int j0 = 0; j0 < 256; j0 += 32) {
        // ---- stage A: gathered pixel tile, 64 b x 32 j (coalesced in j) ----
        for (int e = tid; e < 2048; e += 128) {
            int bb = e >> 5, jj = e & 31;
            int j = j0 + jj;
            float v = 0.0f;
            if (j < jlen) {
                int pix = is_eq ? (EQ_START + (k - RING_EQ0) * NLON + j)
                                : cap_idx[cr * LCAP + j];
                v = x[(size_t)(b0 + bb) * NPIX + pix];
            }
            As[jj][bb] = v;
        }
        // ---- stage B: DFT coefficients, 16 m x 32 j ----
        for (int e = tid; e < 512; e += 128) {
            int mm = e >> 5, jj = e & 31;
            int j = j0 + jj;
            int m = m0 + mm;
            float vr = 0.0f, vi = 0.0f;
            if (is_eq) {
                vr = Fre[m * NLON + j];
                vi = Fim[m * NLON + j];
            } else if (j < LCAP && m < MMAX) {
                const float* e2 = &E_cap[(((size_t)cr * MMAX + m) * LCAP + j) * 2];
                vr = e2[0]; vi = e2[1];
            }
            Bre[jj][mm] = vr;
            Bim[jj][mm] = vi;
        }
        __syncthreads();
        // ---- 8 WMMA K=4 steps; re/im share the A fragment ----
        for (int kk = 0; kk < 8; ++kk) {
            int kb = kk * 4;
            v2f a, br, bi;
            a.x  = As [kb + koff0    ][wb + row];
            a.y  = As [kb + koff0 + 1][wb + row];
            br.x = Bre[kb + koff0    ][row];
            br.y = Bre[kb + koff0 + 1][row];
            bi.x = Bim[kb + koff0    ][row];
            bi.y = Bim[kb + koff0 + 1][row];
            acc_re = wmma_f32(a, br, acc_re);
            acc_im = wmma_f32(a, bi, acc_im);
        }
        __syncthreads();
    }

    // ---- apply equatorial ring phase, store c[k][m][b] ----
    int m_g = m0 + (lane & 15);           // C/D: N = lane&15
    if (m_g < MMAX) {
        float pr = 1.0f, pi = 0.0f;
        if (is_eq) {
            const float* p = &eq_phase[((k - RING_EQ0) * MMAX + m_g) * 2];
            pr = p[0]; pi = p[1];
        }
        int bbase = (lane < 16) ? 0 : 8;  // C/D: M = r (+8 for upper lanes)
        size_t base = ((size_t)k * MPAD + m_g) * BT + b0 + wb + bbase;
        for (int r = 0; r < 8; ++r) {
            float ar = acc_re[r], ai = acc_im[r];
            c_re[base + r] = ar * pr - ai * pi;
            c_im[base + r] = ar * pi + ai * pr;
        }
    }
}

// ---------------------------------------------------------------------
// Phase 2: Legendre contraction. out[b][l][m] = sum_k c[b,k,m] * W[m,l,k]
// Block = 128 threads (4 waves) covering 16 b x 64 l for one m:
// the HBM-resident c tile is staged once, each wave computes a 16-l slice.
// grid = (MMAX, BT/16, LMAX_supertiles=4)
// ---------------------------------------------------------------------
__global__ void __launch_bounds__(128) sht_phase2(
    const float* __restrict__ c_re, const float* __restrict__ c_im,
    const float* __restrict__ W, float2* __restrict__ out)
{
    const int m    = blockIdx.x;          // 0..128
    const int b0   = blockIdx.y * 16;
    const int l0   = blockIdx.z * 64;
    const int tid  = threadIdx.x;
    const int w    = tid >> 5;
    const int lane = tid & 31;

    __shared__ float Ar[32][PITCH16];     // [k][b(16)]
    __shared__ float Ai[32][PITCH16];
    __shared__ float Bw[32][PITCH64];     // [k][l(64)]

    v8f acc_re = {}; v8f acc_im = {};
    const int row   = lane & 15;
    const int koff0 = (lane < 16) ? 0 : 2;
    const int wl    = w * 16;             // wave's l offset inside the tile

    const float* Wm = W + (size_t)m * LMAX * NLAT;

    for (int k0 = 0; k0 < 256; k0 += 32) {
        // ---- stage A: c tiles (re & im), 16 b x 32 k (coalesced in b) ----
        for (int e = tid; e < 512; e += 128) {
            int kk = e >> 4, bb = e & 15;
            int kg = k0 + kk;
            float vr = 0.0f, vi = 0.0f;
            if (kg < NLAT) {
                size_t idx = ((size_t)kg * MPAD + m) * BT + b0 + bb;
                vr = c_re[idx]; vi = c_im[idx];
            }
            Ar[kk][bb] = vr;
            Ai[kk][bb] = vi;
        }
        // ---- stage B: weights (transposed read), 64 l x 32 k ----
        for (int e = tid; e < 2048; e += 128) {
            int ll = e >> 5, kk = e & 31;
            int kg = k0 + kk, lg = l0 + ll;
            float v = 0.0f;
            if (kg < NLAT && lg < LMAX) v = Wm[(size_t)lg * NLAT + kg];
            Bw[kk][ll] = v;
        }
        if (k0 + 32 < 256)   // CDNA5 global_prefetch_b8 on next weight chunk
            __builtin_prefetch(&Wm[(size_t)(l0 + (tid >> 1)) * NLAT + k0 + 32], 0, 0);
        __syncthreads();
        // ---- 8 WMMA K=4 steps; re/im share the B (weights) fragment ----
        for (int kk = 0; kk < 8; ++kk) {
            int kb = kk * 4;
            v2f ar, ai, b;
            ar.x = Ar[kb + koff0][row];       ar.y = Ar[kb + koff0 + 1][row];
            ai.x = Ai[kb + koff0][row];       ai.y = Ai[kb + koff0 + 1][row];
            b.x  = Bw[kb + koff0][wl + row];  b.y  = Bw[kb + koff0 + 1][wl + row];
            acc_re = wmma_f32(ar, b, acc_re);
            acc_im = wmma_f32(ai, b, acc_im);
        }
        __syncthreads();
    }

    int l_g = l0 + wl + (lane & 15);
    if (l_g < LMAX) {
        int bbase = b0 + ((lane < 16) ? 0 : 8);
        for (int r = 0; r < 8; ++r) {
            size_t o = ((size_t)(bbase + r) * LMAX + l_g) * MMAX + m;
            out[o] = make_float2(acc_re[r], acc_im[r]);
        }
    }
}

// ---------------------------------------------------------------------
extern "C" void kernel_launch(void* const* d_in, const int* in_sizes, int n_in,
                              void* d_out, int out_size, void* d_ws, size_t ws_size,
                              hipStream_t stream)
{
    (void)in_sizes; (void)n_in; (void)out_size; (void)ws_size;
    const float* x       = (const float*)d_in[0];   // [8,64,NPIX] f32
    const float* weights = (const float*)d_in[1];   // [MMAX,LMAX,NLAT] f32
    const float* E_cap   = (const float*)d_in[2];   // [126,MMAX,LCAP] c64 (interleaved)
    const float* eq_ph   = (const float*)d_in[3];   // [NEQ,MMAX] c64 (interleaved)
    const int*   cap_idx = (const int*)d_in[4];     // [126,LCAP] i32

    // workspace layout (~138 MB):
    const size_t CPLANE = (size_t)NLAT * MPAD * BT; // 17,233,920 floats
    float* c_re = (float*)d_ws;
    float* c_im = c_re + CPLANE;
    float* Fre  = c_im + CPLANE;
    float* Fim  = Fre + (size_t)FM * NLON;

    build_dft_kernel<<<dim3(FM), dim3(NLON), 0, stream>>>(Fre, Fim);

    dim3 g1(NLAT, BT / 64, (MMAX + 15) / 16);
    sht_phase1<<<g1, dim3(128), 0, stream>>>(x, E_cap, eq_ph, cap_idx,
                                             Fre, Fim, c_re, c_im);

    dim3 g2(MMAX, BT / 16, (LMAX + 63) / 64);
    sht_phase2<<<g2, dim3(128), 0, stream>>>(c_re, c_im, weights, (float2*)d_out);
}